// FocalLoss_89386859364489
// MI455X (gfx1250) — compile-verified
//
#include <hip/hip_runtime.h>

typedef __attribute__((ext_vector_type(2))) float v2f;
typedef __attribute__((ext_vector_type(4))) float v4f;
typedef __attribute__((ext_vector_type(8))) float v8f;

#define FL_EPS 1e-7f

// Per-element focal loss, branchless (t is exactly 0.0 or 1.0).
//  p   = clamp(x, eps, 1-eps)
//  bce = -(t*log(p) + (1-t)*log(1-p))
//  pt  = t ? p : (1-p)
//  w   = t ? 0.8 : 0.2
//  loss = w * (1-pt)^2 * bce          (GAMMA = 2)
__device__ __forceinline__ float focal_term(float x, float t) {
    float p     = fminf(fmaxf(x, FL_EPS), 1.0f - FL_EPS);
    float omp   = 1.0f - p;
    float logp  = __logf(p);
    float logmp = __logf(omp);
    float bce   = -fmaf(t, logp - logmp, logmp);
    float pt    = fmaf(t, p - omp, omp);
    float ompt  = 1.0f - pt;
    float w     = fmaf(t, 0.6f, 0.2f);
    return w * ompt * ompt * bce;
}

// Wave32 sum-reduction via V_WMMA_F32_16X16X4_F32.
// A[m][0] = partial(lane m), A[m][2] = partial(lane m+16), K=1,3 slots zero.
// B = ones(4x16)  =>  D[m][n] = partial[m] + partial[m+16] for every n.
// D vgpr r: lanes 0-15 hold row r, lanes 16-31 hold row r+8, so summing the
// 8 accumulator VGPRs gives rows 0-7 (low lanes) / 8-15 (high lanes); one
// xor-16 shuffle completes the 32-lane sum. Requires EXEC == all 1s.
__device__ __forceinline__ float wave_reduce_sum(float partial) {
    v2f a; a[0] = partial; a[1] = 0.0f;
    v2f b; b[0] = 1.0f;    b[1] = 1.0f;
    v8f c = {};
    v8f d = __builtin_amdgcn_wmma_f32_16x16x4_f32(
        /*neg_a=*/false, a, /*neg_b=*/false, b,
        /*c_mod=*/(short)0, c, /*reuse_a=*/false, /*reuse_b=*/false);
    float s = ((d[0] + d[1]) + (d[2] + d[3])) + ((d[4] + d[5]) + (d[6] + d[7]));
    s += __shfl_xor(s, 16, 32);
    return s;
}

// Block-level reduction; result valid in threadIdx.x == 0.
__device__ __forceinline__ float block_reduce_sum(float v) {
    __shared__ float lds[32];
    float s   = wave_reduce_sum(v);
    int lane  = threadIdx.x & 31;
    int wid   = threadIdx.x >> 5;
    if (lane == 0) lds[wid] = s;
    __syncthreads();
    float tot = 0.0f;
    if (threadIdx.x == 0) {
        int nw = (blockDim.x + 31) >> 5;
        for (int i = 0; i < nw; ++i) tot += lds[i];
    }
    return tot;
}

// Pass 1: streaming focal-loss + per-block partial sums into d_ws.
__global__ void __launch_bounds__(256)
focal_partial_kernel(const float* __restrict__ in, const float* __restrict__ tg,
                     float* __restrict__ ws, long long n) {
    long long tid    = (long long)blockIdx.x * blockDim.x + threadIdx.x;
    long long stride = (long long)gridDim.x * blockDim.x;

    float acc = 0.0f;

    // Vectorized body: b128 non-temporal streaming loads.
    long long nv = n >> 2;
    const v4f* __restrict__ in4 = (const v4f*)in;
    const v4f* __restrict__ tg4 = (const v4f*)tg;
    for (long long i = tid; i < nv; i += stride) {
        v4f x = __builtin_nontemporal_load(&in4[i]);
        v4f t = __builtin_nontemporal_load(&tg4[i]);
        acc += focal_term(x[0], t[0]);
        acc += focal_term(x[1], t[1]);
        acc += focal_term(x[2], t[2]);
        acc += focal_term(x[3], t[3]);
    }
    // Scalar tail (reconverges before the WMMA wave reduction).
    for (long long i = (nv << 2) + tid; i < n; i += stride) {
        acc += focal_term(in[i], tg[i]);
    }

    float tot = block_reduce_sum(acc);
    if (threadIdx.x == 0) ws[blockIdx.x] = tot;
}

// Pass 2: single block reduces the per-block partials -> d_out[0].
// Fully deterministic (fixed summation order, no atomics).
__global__ void __launch_bounds__(256)
focal_final_kernel(const float* __restrict__ ws, float* __restrict__ out, int m) {
    float acc = 0.0f;
    for (int i = threadIdx.x; i < m; i += blockDim.x) acc += ws[i];
    float tot = block_reduce_sum(acc);
    if (threadIdx.x == 0) out[0] = tot;
}

extern "C" void kernel_launch(void* const* d_in, const int* in_sizes, int n_in,
                              void* d_out, int out_size, void* d_ws, size_t ws_size,
                              hipStream_t stream) {
    const float* inputs  = (const float*)d_in[0];
    const float* targets = (const float*)d_in[1];
    float*       out     = (float*)d_out;
    float*       ws      = (float*)d_ws;
    long long    n       = (long long)in_sizes[0];

    // 1024 blocks x 256 threads = 8192 wave32s: enough in-flight b128 loads
    // to saturate 23.3 TB/s. Defensive cap against a tiny workspace.
    int blocks = 1024;
    long long ws_cap = (long long)(ws_size / sizeof(float));
    if (ws_cap > 0 && ws_cap < blocks) blocks = (int)ws_cap;
    if (blocks < 1) blocks = 1;

    focal_partial_kernel<<<blocks, 256, 0, stream>>>(inputs, targets, ws, n);
    focal_final_kernel<<<1, 256, 0, stream>>>(ws, out, blocks);
}